// SoftMinLayer_54597624267331
// MI455X (gfx1250) — compile-verified
//
#include <hip/hip_runtime.h>
#include <stdint.h>

typedef __attribute__((ext_vector_type(2))) float v2f;
typedef __attribute__((ext_vector_type(8))) float v8f;

#define ALPHA_T (-100.0f)

constexpr int Q      = 2048;
constexpr int QPAD   = Q + 64;          // zero-padded so no clamping needed anywhere
constexpr int L      = 50;
constexpr int J      = Q - L + 1;       // 1999
constexpr int LP     = 52;              // L padded to multiple of 4
constexpr int KSTEPS = LP / 4;          // 13 WMMA k-steps
constexpr int JTILES = (J + 15) / 16;   // 125
constexpr int BLOCK  = 128;             // 4 waves (wave32)
constexpr int WAVES  = BLOCK / 32;

// Large-but-safe sentinel: D becomes ~2e35 for masked columns; fminf ignores it
// and __expf(-100 * 2e35) == 0 so softmin sums get exact-zero contributions.
#define MASK_BIG 1.0e37f

// Monotone mapping float -> uint so unsigned atomicMin == float min (handles negatives)
__device__ __forceinline__ unsigned enc_f32(float f) {
    unsigned u = __float_as_uint(f);
    return (u & 0x80000000u) ? ~u : (u | 0x80000000u);
}
__device__ __forceinline__ float dec_f32(unsigned u) {
    unsigned b = (u & 0x80000000u) ? (u ^ 0x80000000u) : ~u;
    return __uint_as_float(b);
}

__global__ void softmin_init_min(unsigned* __restrict__ minD, int K) {
    int t = blockIdx.x * blockDim.x + threadIdx.x;
    if (t < K) minD[t] = 0xFF800000u;   // enc(+inf)
}

template <bool MINPASS>
__global__ __launch_bounds__(BLOCK)
void softmin_pass(const float* __restrict__ X, const float* __restrict__ S,
                  unsigned* __restrict__ minD, float* __restrict__ out,
                  int n, int K)
{
    __shared__ float xs[QPAD];       // one series row + zero pad
    __shared__ float sw[2000];       // sliding-window sum of squares
    __shared__ float ssq[16];        // shapelet row norms for this k-tile
    __shared__ float num_s[WAVES][16];
    __shared__ float den_s[WAVES][16];

    const int nkt   = K >> 4;
    const int i     = blockIdx.x / nkt;
    const int k0    = (blockIdx.x % nkt) << 4;
    const int tid   = threadIdx.x;
    const int lane  = tid & 31;
    const int wave  = tid >> 5;
    const int laneN = lane & 15;     // N (j) index within tile / M row for A
    const int laneH = lane >> 4;     // half-wave selector

    // ---- stage series row to LDS (vectorized b128 loads) + zero pad ----
    const float4* Xv = (const float4*)(X + (size_t)i * Q);
    for (int v = tid; v < Q / 4; v += BLOCK) {
        float4 d = Xv[v];
        xs[v * 4 + 0] = d.x; xs[v * 4 + 1] = d.y;
        xs[v * 4 + 2] = d.z; xs[v * 4 + 3] = d.w;
    }
    if (tid < QPAD - Q) xs[Q + tid] = 0.f;
    __syncthreads();

    // ---- window sum-of-squares for every j (pad makes all reads in-bounds) ----
    for (int j = tid; j < 2000; j += BLOCK) {
        float acc = 0.f;
        #pragma unroll 2
        for (int l = 0; l < L; ++l) {
            float v = xs[j + l];
            acc = fmaf(v, v, acc);
        }
        sw[j] = acc;
    }
    // ---- shapelet row norms for this k-tile ----
    if (tid < 16) {
        const float* srow = S + (size_t)(k0 + tid) * L;
        float a = 0.f;
        for (int l = 0; l < L; ++l) a = fmaf(srow[l], srow[l], a);
        ssq[tid] = a;
    }
    __syncthreads();

    // ---- A fragments: 16x4 f32 tiles of the shapelet block, zero-padded to LP ----
    // ISA layout: lanes 0-15 hold K={0,1}, lanes 16-31 hold K={2,3}; M = lane&15
    v2f afrag[KSTEPS];
    {
        const float* srow = S + (size_t)(k0 + laneN) * L;
        #pragma unroll
        for (int s = 0; s < KSTEPS; ++s) {
            int c = s * 4 + laneH * 2;
            v2f a;
            a[0] = (c     < L) ? srow[c]     : 0.f;
            a[1] = (c + 1 < L) ? srow[c + 1] : 0.f;
            afrag[s] = a;
        }
    }

    float ssq_r[8];
    #pragma unroll
    for (int r = 0; r < 8; ++r) ssq_r[r] = ssq[r + laneH * 8];

    float mk[8];
    if (!MINPASS) {
        #pragma unroll
        for (int r = 0; r < 8; ++r) mk[r] = dec_f32(minD[k0 + r + laneH * 8]);
    }

    float minv[8], num[8], den[8];
    #pragma unroll
    for (int r = 0; r < 8; ++r) { minv[r] = 3.0e38f; num[r] = 0.f; den[r] = 0.f; }

    // ---- j-tile sweep for this wave ----
    const int per = (JTILES + WAVES - 1) / WAVES;
    const int t0  = wave * per;
    const int t1  = (t0 + per < JTILES) ? (t0 + per) : JTILES;
    const float invL = 1.0f / (float)L;

    for (int jt = t0; jt < t1; ++jt) {
        const int jlane = jt * 16 + laneN;
        // branchless column mask: bogus columns get a huge window norm
        const float swv = (jlane < J) ? sw[jlane] : MASK_BIG;

        // gather all B fragments from one base with immediate offsets, then one
        // wait and 13 back-to-back WMMAs (accumulate chaining has no hazard)
        const float* bp = &xs[jlane + laneH * 2];
        v2f bfrag[KSTEPS];
        #pragma unroll
        for (int s = 0; s < KSTEPS; ++s) {
            v2f b; b[0] = bp[s * 4]; b[1] = bp[s * 4 + 1];
            bfrag[s] = b;
        }

        v8f acc = {};
        #pragma unroll
        for (int s = 0; s < KSTEPS; ++s) {
            acc = __builtin_amdgcn_wmma_f32_16x16x4_f32(
                false, afrag[s], false, bfrag[s], (short)0, acc, false, false);
        }

        #pragma unroll
        for (int r = 0; r < 8; ++r) {
            float D = (swv - 2.0f * acc[r] + ssq_r[r]) * invL;
            if (MINPASS) {
                minv[r] = fminf(minv[r], D);
            } else {
                float Ds = D - mk[r];
                float e  = __expf(ALPHA_T * Ds);   // masked cols: e == 0 exactly
                num[r] = fmaf(Ds, e, num[r]);
                den[r] += e;
            }
        }
    }

    if (MINPASS) {
        // per-k min: reduce within each 16-lane half, then global atomicMin
        #pragma unroll
        for (int r = 0; r < 8; ++r) {
            float m = minv[r];
            m = fminf(m, __shfl_xor(m, 1, 32));
            m = fminf(m, __shfl_xor(m, 2, 32));
            m = fminf(m, __shfl_xor(m, 4, 32));
            m = fminf(m, __shfl_xor(m, 8, 32));
            if (laneN == 0) atomicMin(&minD[k0 + r + laneH * 8], enc_f32(m));
        }
    } else {
        // deterministic cross-wave reduction: fixed-order LDS partials (no float atomics)
        #pragma unroll
        for (int r = 0; r < 8; ++r) {
            float nu = num[r], de = den[r];
            nu += __shfl_xor(nu, 1, 32); de += __shfl_xor(de, 1, 32);
            nu += __shfl_xor(nu, 2, 32); de += __shfl_xor(de, 2, 32);
            nu += __shfl_xor(nu, 4, 32); de += __shfl_xor(de, 4, 32);
            nu += __shfl_xor(nu, 8, 32); de += __shfl_xor(de, 8, 32);
            if (laneN == 0) {
                num_s[wave][r + laneH * 8] = nu;
                den_s[wave][r + laneH * 8] = de;
            }
        }
        __syncthreads();
        if (tid < 16) {
            float nu = 0.f, de = 0.f;
            #pragma unroll
            for (int w = 0; w < WAVES; ++w) { nu += num_s[w][tid]; de += den_s[w][tid]; }
            out[(size_t)i * K + k0 + tid] = nu / de;
        }
    }
}

extern "C" void kernel_launch(void* const* d_in, const int* in_sizes, int n_in,
                              void* d_out, int out_size, void* d_ws, size_t ws_size,
                              hipStream_t stream) {
    const float* X = (const float*)d_in[0];   // (n, Q)
    const float* S = (const float*)d_in[1];   // (K, L)
    float* out = (float*)d_out;               // (n, K)
    const int n = in_sizes[0] / Q;            // 512
    const int K = in_sizes[1] / L;            // 64
    unsigned* minD = (unsigned*)d_ws;         // 64 * 4 bytes

    softmin_init_min<<<1, 64, 0, stream>>>(minD, K);
    dim3 grid(n * (K / 16));
    softmin_pass<true ><<<grid, BLOCK, 0, stream>>>(X, S, minD, out, n, K);
    softmin_pass<false><<<grid, BLOCK, 0, stream>>>(X, S, minD, out, n, K);
}